// Decoder_DIBS_21818433864251
// MI455X (gfx1250) — compile-verified
//
#include <hip/hip_runtime.h>
#include <hip/hip_bf16.h>
#include <math.h>

// ---------------- problem constants (from reference) ----------------
#define MPART 256      // particles
#define DNODE 64       // nodes
#define KLAT  32       // latent rank
#define NOBS  2048     // observations
#define ROWS  8450     // out row stride: 256 + 4096 + 4096 + 1 + 1
#define H_LATENT 5.0f
#define P_EDGE   (64.0f / 2016.0f)          // EDGES_PER_NODE*D / (D*(D-1)/2)
#define N_PAIRS  2016.0f
#define SMALL_T  15.5f                      // alpha_mu*(alpha_lambd-d-1)/(alpha_mu+1)
#define N_PLUS_AMU 2049.0f                  // N + alpha_mu
#define PI_F 3.14159265358979323846f

typedef __attribute__((ext_vector_type(2))) float v2f;
typedef __attribute__((ext_vector_type(8))) float v8f;

// ---------------------------------------------------------------
// 1) sq[m] = ||zf[m]||^2 over 4096 elems
// ---------------------------------------------------------------
__global__ void sq_kernel(const float* __restrict__ z, float* __restrict__ sq) {
  int m = blockIdx.x, tid = threadIdx.x;
  const float* row = z + (size_t)m * 4096;
  float acc = 0.0f;
  for (int k = tid; k < 4096; k += 256) { float v = row[k]; acc += v * v; }
  __shared__ float red[256];
  red[tid] = acc; __syncthreads();
  for (int s = 128; s > 0; s >>= 1) { if (tid < s) red[tid] += red[tid + s]; __syncthreads(); }
  if (tid == 0) sq[m] = red[0];
}

// ---------------------------------------------------------------
// 2) kxx: fp32 WMMA GEMM zf @ zf^T (K=4096) fused with RBF epilogue.
//    grid (16,16) tiles of 16x16, one wave (32 lanes) per tile.
// ---------------------------------------------------------------
__global__ void kxx_kernel(const float* __restrict__ z, const float* __restrict__ sq,
                           float* __restrict__ out) {
  int tc = blockIdx.x, tr = blockIdx.y;
  int lane = threadIdx.x;                 // 0..31
  int mrow = tr * 16 + (lane & 15);       // A-frag row (lanes 0-15 and 16-31 both hold M=0..15)
  int ncol = tc * 16 + (lane & 15);       // B-frag col
  int koff = (lane >> 4) * 2;             // lanes 16-31 hold K+2,K+3
  const float* arow = z + (size_t)mrow * 4096;
  const float* brow = z + (size_t)ncol * 4096;
  v8f acc = {};
  for (int k0 = 0; k0 < 4096; k0 += 4) {
    v2f a, b;
    a.x = arow[k0 + koff];  a.y = arow[k0 + koff + 1];
    b.x = brow[k0 + koff];  b.y = brow[k0 + koff + 1];
    acc = __builtin_amdgcn_wmma_f32_16x16x4_f32(false, a, false, b, (short)0, acc,
                                                false, false);
  }
  int rbase = tr * 16 + ((lane >> 4) << 3);
  int col   = tc * 16 + (lane & 15);
  for (int r = 0; r < 8; ++r) {
    int row = rbase + r;
    float d2 = sq[row] + sq[col] - 2.0f * acc[r];
    out[(size_t)row * ROWS + col] = expf(-d2 / H_LATENT);   // SCALE = 1
  }
}

// ---------------------------------------------------------------
// 3) scores = U V^T per particle (64x64, K=32) via fp32 WMMA,
//    fused log_sigmoid / sigmoid epilogue + log_prior reduction.
//    one block of 512 threads (16 waves = 16 tiles) per particle.
// ---------------------------------------------------------------
__global__ void scores_kernel(const float* __restrict__ z, const int* __restrict__ tptr,
                              float* __restrict__ out) {
  int m = blockIdx.x;
  int tidx = threadIdx.x;
  int wave = tidx >> 5, lane = tidx & 31;
  int tr = wave >> 2, tc = wave & 3;
  float alpha = 0.05f * (float)tptr[0];
  const float* zm = z + (size_t)m * DNODE * KLAT * 2;
  // u[i][k] = zm[(i*32+k)*2], v[j][k] = zm[(j*32+k)*2 + 1]
  int ai = tr * 16 + (lane & 15);
  int bj = tc * 16 + (lane & 15);
  int koff = (lane >> 4) * 2;
  v8f acc = {};
  for (int k0 = 0; k0 < KLAT; k0 += 4) {
    v2f a, b;
    a.x = zm[(ai * KLAT + k0 + koff) * 2];
    a.y = zm[(ai * KLAT + k0 + koff + 1) * 2];
    b.x = zm[(bj * KLAT + k0 + koff) * 2 + 1];
    b.y = zm[(bj * KLAT + k0 + koff + 1) * 2 + 1];
    acc = __builtin_amdgcn_wmma_f32_16x16x4_f32(false, a, false, b, (short)0, acc,
                                                false, false);
  }
  float psum = 0.0f;
  int rbase = tr * 16 + ((lane >> 4) << 3);
  int col   = tc * 16 + (lane & 15);
  float* orow = out + (size_t)m * ROWS;
  for (int r = 0; r < 8; ++r) {
    int row = rbase + r;
    float lp, lpn, p;
    if (row == col) { lp = 0.0f; lpn = 0.0f; p = 0.0f; }
    else {
      float x = alpha * acc[r];
      if (x >= 0.0f) { float e = expf(-x); lp = -log1pf(e); lpn = lp - x; p = 1.0f / (1.0f + e); }
      else           { float e = expf(x);  lpn = -log1pf(e); lp = lpn + x; p = e / (1.0f + e); }
    }
    orow[256  + row * 64 + col] = lp;
    orow[4352 + row * 64 + col] = lpn;
    psum += p;
  }
  __shared__ float red[512];
  red[tidx] = psum; __syncthreads();
  for (int s = 256; s > 0; s >>= 1) { if (tidx < s) red[tidx] += red[tidx + s]; __syncthreads(); }
  if (tidx == 0) {
    float e = red[0];
    orow[8449] = e * logf(P_EDGE) + (N_PAIRS - e) * logf(1.0f - P_EDGE);
  }
}

// ---------------------------------------------------------------
// 4) column sums of data (for the mean / rank-1 correction of R)
// ---------------------------------------------------------------
__global__ void colsum_kernel(const float* __restrict__ data, float* __restrict__ s) {
  int j = threadIdx.x;   // 64 threads
  float acc = 0.0f;
  for (int n = 0; n < NOBS; ++n) acc += data[(size_t)n * DNODE + j];
  s[j] = acc;
}

// ---------------------------------------------------------------
// 5) R = small_t*I + data^T data - (s s^T)/(N+alpha_mu)
//    (xc^T xc + N*amu/(N+amu) xbar^T xbar simplified exactly)
//    one block, 512 threads = 16 waves = 16 tiles, K=2048 fp32 WMMA.
// ---------------------------------------------------------------
__global__ void R_kernel(const float* __restrict__ data, const float* __restrict__ s,
                         float* __restrict__ R) {
  int tidx = threadIdx.x;
  int wave = tidx >> 5, lane = tidx & 31;
  int tr = wave >> 2, tc = wave & 3;
  int i = tr * 16 + (lane & 15);
  int j = tc * 16 + (lane & 15);
  int koff = (lane >> 4) * 2;
  v8f acc = {};
  for (int k0 = 0; k0 < NOBS; k0 += 4) {
    v2f a, b;
    a.x = data[(size_t)(k0 + koff) * DNODE + i];
    a.y = data[(size_t)(k0 + koff + 1) * DNODE + i];
    b.x = data[(size_t)(k0 + koff) * DNODE + j];
    b.y = data[(size_t)(k0 + koff + 1) * DNODE + j];
    acc = __builtin_amdgcn_wmma_f32_16x16x4_f32(false, a, false, b, (short)0, acc,
                                                false, false);
  }
  int rbase = tr * 16 + ((lane >> 4) << 3);
  int col   = tc * 16 + (lane & 15);
  for (int r = 0; r < 8; ++r) {
    int row = rbase + r;
    float val = acc[r] - s[row] * s[col] * (1.0f / N_PLUS_AMU);
    if (row == col) val += SMALL_T;
    R[row * DNODE + col] = val;
  }
}

// ---------------------------------------------------------------
// 6) BGe node scores: masked-identity Cholesky logdets.
//    One 64-thread block per (m, j). Thread == matrix row.
// ---------------------------------------------------------------
__device__ float masked_logdet(float (*A)[65], const int* par, const float* __restrict__ R,
                               int tid, int extra) {
  bool mi = (par[tid] != 0) || (tid == extra);
  for (int b = 0; b < DNODE; ++b) {
    bool mb = (par[b] != 0) || (b == extra);
    A[tid][b] = (mi && mb) ? R[tid * DNODE + b] : (tid == b ? 1.0f : 0.0f);
  }
  __syncthreads();
  float ld = 0.0f;
  for (int k = 0; k < DNODE; ++k) {
    float dk = A[k][k];          // updated diag = Lkk^2
    ld += logf(dk);
    float invL = 1.0f / sqrtf(dk);
    if (tid > k) A[tid][k] *= invL;
    __syncthreads();
    if (tid > k) {
      float lik = A[tid][k];
      for (int b = k + 1; b <= tid; ++b) A[tid][b] -= lik * A[b][k];
    }
    __syncthreads();
  }
  return ld;                     // sum log(dk) == log det
}

__device__ float bge_log_gamma(int np) {
  float f = (float)np;
  return -0.5f * logf(N_PLUS_AMU)
       + lgammaf(0.5f * (2081.0f + f))     // 0.5*(N + aL - d + np + 1)
       - lgammaf(0.5f * (33.0f + f))       // 0.5*(aL - d + np + 1)
       - 1024.0f * logf(PI_F)              // 0.5*N*log(pi)
       + 0.5f * (33.0f + 2.0f * f) * logf(SMALL_T);
}

__global__ void bge_node_kernel(const int* __restrict__ gs, const float* __restrict__ R,
                                float* __restrict__ node_scores) {
  int j = blockIdx.x;      // node
  int m = blockIdx.y;      // particle
  int tid = threadIdx.x;   // 0..63
  __shared__ float A[DNODE][65];
  __shared__ int par[DNODE];
  par[tid] = (gs[(size_t)m * DNODE * DNODE + tid * DNODE + j] != 0) ? 1 : 0;
  __syncthreads();
  int np = 0;
  for (int i = 0; i < DNODE; ++i) np += par[i];
  float ld1 = 0.0f, ld2 = 0.0f;
  if (np > 0) {
    ld1 = masked_logdet(A, par, R, tid, -1);   // parents
    ld2 = masked_logdet(A, par, R, tid, j);    // parents + j
  }
  if (tid == 0) {
    float lg = bge_log_gamma(np);
    float score;
    if (np == 0) {
      score = lg - 0.5f * 2081.0f * logf(fabsf(R[j * DNODE + j]));
    } else {
      score = lg + 0.5f * (2080.0f + (float)np) * ld1
                 - 0.5f * (2081.0f + (float)np) * ld2;
    }
    node_scores[m * DNODE + j] = score;
  }
}

// ---------------------------------------------------------------
// 7) deterministic reduction of node scores -> log_liks column
// ---------------------------------------------------------------
__global__ void reduce_ll_kernel(const float* __restrict__ ns, float* __restrict__ out) {
  int m = blockIdx.x, tid = threadIdx.x;
  __shared__ float red[DNODE];
  red[tid] = ns[m * DNODE + tid]; __syncthreads();
  for (int s = 32; s > 0; s >>= 1) { if (tid < s) red[tid] += red[tid + s]; __syncthreads(); }
  if (tid == 0) out[(size_t)m * ROWS + 8448] = red[0];
}

// ---------------------------------------------------------------
extern "C" void kernel_launch(void* const* d_in, const int* in_sizes, int n_in,
                              void* d_out, int out_size, void* d_ws, size_t ws_size,
                              hipStream_t stream) {
  const float* z    = (const float*)d_in[0];   // [256,64,32,2]
  const float* data = (const float*)d_in[1];   // [2048,64]
  const int*   gs   = (const int*)d_in[2];     // [256,64,64]
  const int*   t    = (const int*)d_in[3];     // scalar
  float* out = (float*)d_out;                  // [256, 8450]

  float* wsf        = (float*)d_ws;
  float* sq         = wsf;                     // 256
  float* s          = wsf + 256;               // 64
  float* R          = wsf + 512;               // 64*64
  float* node_sc    = wsf + 4608;              // 256*64

  sq_kernel      <<<MPART, 256, 0, stream>>>(z, sq);
  kxx_kernel     <<<dim3(16, 16), 32, 0, stream>>>(z, sq, out);
  scores_kernel  <<<MPART, 512, 0, stream>>>(z, t, out);
  colsum_kernel  <<<1, DNODE, 0, stream>>>(data, s);
  R_kernel       <<<1, 512, 0, stream>>>(data, s, R);
  bge_node_kernel<<<dim3(DNODE, MPART), DNODE, 0, stream>>>(gs, R, node_sc);
  reduce_ll_kernel<<<MPART, DNODE, 0, stream>>>(node_sc, out);
  (void)in_sizes; (void)n_in; (void)out_size; (void)ws_size;
}